// Select_76682346102870
// MI455X (gfx1250) — compile-verified
//
#include <hip/hip_runtime.h>

typedef __attribute__((ext_vector_type(2))) float v2f;
typedef __attribute__((ext_vector_type(8))) float v8f;

#define BB 128
#define SS 1024
#define HH 256
#define TT 10
#define NEG_INF -1e8f

// ---------------------------------------------------------------------------
// Generic fp32 WMMA GEMM:  C[M,N] = A[M,K] * W[N,K]^T + bias[N]
// K is a compile-time power of two (=256 for every GEMM here).
// Requires M%16==0, N%64==0. One wave computes a 16(M) x 64(N) strip.
// Software-pipelined: fragments for k+4 are loaded before the k WMMAs issue,
// so the compiler can wait on loadcnt<=5 instead of loadcnt==0.
// ---------------------------------------------------------------------------
template <int K>
__global__ void gemm_wmma_f32(const float* __restrict__ A,
                              const float* __restrict__ W,
                              const float* __restrict__ bias,
                              float* __restrict__ C,
                              int M, int N) {
    const int wave = (blockIdx.x * blockDim.x + threadIdx.x) >> 5;
    const int lane = threadIdx.x & 31;
    const int strips = N >> 6;             // N/64
    const int m_tile = wave / strips;
    const int strip  = wave % strips;
    if (m_tile * 16 >= M) return;          // wave-uniform exit (EXEC stays all-1s)
    const int m0 = m_tile * 16;
    const int n0 = strip * 64;
    const int l  = lane & 15;
    const int hi = lane >> 4;              // 0 -> K pair {0,1}; 1 -> {2,3}

    // Accumulators preloaded with bias (same column for all 8 rows in a lane).
    v8f c0, c1, c2, c3;
    const float b0v = bias[n0 +  0 + l];
    const float b1v = bias[n0 + 16 + l];
    const float b2v = bias[n0 + 32 + l];
    const float b3v = bias[n0 + 48 + l];
#pragma unroll
    for (int r = 0; r < 8; ++r) { c0[r] = b0v; c1[r] = b1v; c2[r] = b2v; c3[r] = b3v; }

    const float* arow = A + (size_t)(m0 + l) * K + 2 * hi;
    const float* w0r  = W + (size_t)(n0 +  0 + l) * K + 2 * hi;
    const float* w1r  = W + (size_t)(n0 + 16 + l) * K + 2 * hi;
    const float* w2r  = W + (size_t)(n0 + 32 + l) * K + 2 * hi;
    const float* w3r  = W + (size_t)(n0 + 48 + l) * K + 2 * hi;

    // Prologue: load chunk 0.
    v2f a  = *(const v2f*)(arow);
    v2f b0 = *(const v2f*)(w0r);
    v2f b1 = *(const v2f*)(w1r);
    v2f b2 = *(const v2f*)(w2r);
    v2f b3 = *(const v2f*)(w3r);

#pragma unroll 4
    for (int k = 0; k < K; k += 4) {
        // Prefetch next chunk (wraps to 0 on last iteration; always in-bounds).
        const int kn = (k + 4) & (K - 1);
        v2f an  = *(const v2f*)(arow + kn);
        v2f b0n = *(const v2f*)(w0r + kn);
        v2f b1n = *(const v2f*)(w1r + kn);
        v2f b2n = *(const v2f*)(w2r + kn);
        v2f b3n = *(const v2f*)(w3r + kn);

        c0 = __builtin_amdgcn_wmma_f32_16x16x4_f32(false, a, false, b0, (short)0, c0, false, false);
        c1 = __builtin_amdgcn_wmma_f32_16x16x4_f32(false, a, false, b1, (short)0, c1, false, false);
        c2 = __builtin_amdgcn_wmma_f32_16x16x4_f32(false, a, false, b2, (short)0, c2, false, false);
        c3 = __builtin_amdgcn_wmma_f32_16x16x4_f32(false, a, false, b3, (short)0, c3, false, false);

        a = an; b0 = b0n; b1 = b1n; b2 = b2n; b3 = b3n;
    }

    // D layout: VGPR r -> row m0 + r + 8*hi, col n0 + 16*j + l
#pragma unroll
    for (int r = 0; r < 8; ++r) {
        float* crow = C + (size_t)(m0 + r + 8 * hi) * N;
        crow[n0 +  0 + l] = c0[r];
        crow[n0 + 16 + l] = c1[r];
        crow[n0 + 32 + l] = c2[r];
        crow[n0 + 48 + l] = c3[r];
    }
}

// mask[b,s] = (s >= sent_length[b]) ? 1 : 0   -> written to out tail
__global__ void write_mask(const int* __restrict__ sent_length, float* __restrict__ out_mask) {
    int i = blockIdx.x * blockDim.x + threadIdx.x;
    if (i >= BB * SS) return;
    int b = i >> 10;
    int s = i & (SS - 1);
    out_mask[i] = (s >= sent_length[b]) ? 1.0f : 0.0f;
}

// X[0,b,:] = s0[b,:];  X[t,b,:] = doc[b, tgt_idx[b,t-1], :]  (tgt==-1 -> 0)
__global__ void gather_inputs(const float* __restrict__ s0,
                              const float* __restrict__ doc,
                              const int* __restrict__ tgt,
                              float* __restrict__ X) {
    int i = blockIdx.x * blockDim.x + threadIdx.x;
    if (i >= TT * BB * HH) return;
    int t = i / (BB * HH);
    int r = i - t * (BB * HH);
    int b = r >> 8;
    int j = r & (HH - 1);
    float v;
    if (t == 0) {
        v = s0[b * HH + j];
    } else {
        int idx = tgt[b * TT + (t - 1)];
        if (idx == -1) idx = 0;
        v = doc[((size_t)b * SS + idx) * HH + j];
    }
    X[i] = v;
}

// GRU gate fusion: hnew = (1-z)*n + z*hprev
__global__ void gru_update(const float* __restrict__ GIt,   // [B,3H]
                           const float* __restrict__ gh,    // [B,3H]
                           const float* __restrict__ hprev, // [B,H]
                           float* __restrict__ hnew) {      // [B,H]
    int i = blockIdx.x * blockDim.x + threadIdx.x;
    if (i >= BB * HH) return;
    int b = i >> 8;
    int j = i & (HH - 1);
    const float* gi = GIt + (size_t)b * (3 * HH);
    const float* g2 = gh  + (size_t)b * (3 * HH);
    float ir = gi[j], iz = gi[HH + j], in_ = gi[2 * HH + j];
    float hr = g2[j], hz = g2[HH + j], hn  = g2[2 * HH + j];
    float r = 1.0f / (1.0f + expf(-(ir + hr)));
    float z = 1.0f / (1.0f + expf(-(iz + hz)));
    float n = tanhf(in_ + r * hn);
    hnew[i] = (1.0f - z) * n + z * hprev[i];
}

// energy[b,s] = Ws . tanh(pre[b,s,:] + target[b,:]) + bs ; masked -> NEG_INF
// One wave per (b,s); coalesced float4 loads; wave32 shuffle reduction.
__global__ void energy_kernel(const float* __restrict__ pre,    // [B,S,H]
                              const float* __restrict__ target, // [B,H]
                              const float* __restrict__ Ws,     // [H]
                              const float* __restrict__ bs,     // [1]
                              const int* __restrict__ sent_length,
                              float* __restrict__ out) {        // [B,S]
    int wave = (blockIdx.x * blockDim.x + threadIdx.x) >> 5;
    int lane = threadIdx.x & 31;
    if (wave >= BB * SS) return;
    int b = wave >> 10;
    int s = wave & (SS - 1);

    const float4* p  = (const float4*)(pre + (size_t)wave * HH);
    const float4* tg = (const float4*)(target + (size_t)b * HH);
    const float4* w  = (const float4*)Ws;

    float4 p0 = p[lane],  p1 = p[lane + 32];
    float4 t0 = tg[lane], t1 = tg[lane + 32];
    float4 w0 = w[lane],  w1 = w[lane + 32];

    float acc = w0.x * tanhf(p0.x + t0.x) + w0.y * tanhf(p0.y + t0.y)
              + w0.z * tanhf(p0.z + t0.z) + w0.w * tanhf(p0.w + t0.w)
              + w1.x * tanhf(p1.x + t1.x) + w1.y * tanhf(p1.y + t1.y)
              + w1.z * tanhf(p1.z + t1.z) + w1.w * tanhf(p1.w + t1.w);

#pragma unroll
    for (int off = 16; off > 0; off >>= 1) acc += __shfl_xor(acc, off, 32);

    if (lane == 0) {
        float e = acc + bs[0];
        if (s >= sent_length[b]) e = NEG_INF;   // energy*(1-m) + m*NEG_INF with m in {0,1}
        out[wave] = e;
    }
}

extern "C" void kernel_launch(void* const* d_in, const int* in_sizes, int n_in,
                              void* d_out, int out_size, void* d_ws, size_t ws_size,
                              hipStream_t stream) {
    const float* s0   = (const float*)d_in[0];
    const float* h0   = (const float*)d_in[1];
    const float* doc  = (const float*)d_in[2];
    const float* W_ih = (const float*)d_in[3];
    const float* W_hh = (const float*)d_in[4];
    const float* b_ih = (const float*)d_in[5];
    const float* b_hh = (const float*)d_in[6];
    const float* Wq   = (const float*)d_in[7];
    const float* bq   = (const float*)d_in[8];
    const float* Wd   = (const float*)d_in[9];
    const float* bd   = (const float*)d_in[10];
    const float* Ws   = (const float*)d_in[11];
    const float* bs   = (const float*)d_in[12];
    const int*   tgt  = (const int*)d_in[13];
    const int*   slen = (const int*)d_in[14];
    float* out = (float*)d_out;

    // Workspace carve-up (floats): pre is 134 MB and deliberately fits in L2.
    float* ws   = (float*)d_ws;
    float* pre  = ws;                                  // B*S*H
    float* GI   = pre  + (size_t)BB * SS * HH;         // T*B*3H
    float* X    = GI   + (size_t)TT * BB * 3 * HH;     // T*B*H
    float* gh   = X    + (size_t)TT * BB * HH;         // B*3H
    float* tgv  = gh   + (size_t)BB * 3 * HH;          // B*H  (target)
    float* hA   = tgv  + (size_t)BB * HH;              // B*H
    float* hB   = hA   + (size_t)BB * HH;              // B*H

    // mask -> out tail [T*B*S .. T*B*S + B*S)
    write_mask<<<(BB * SS + 255) / 256, 256, 0, stream>>>(slen, out + (size_t)TT * BB * SS);

    // Gather teacher-forced GRU inputs for all steps up-front.
    gather_inputs<<<(TT * BB * HH + 255) / 256, 256, 0, stream>>>(s0, doc, tgt, X);

    // precompute = doc @ Wd^T + bd   (M=131072, N=256, K=256)
    {
        int waves = (BB * SS / 16) * (HH / 64);        // 32768 waves
        gemm_wmma_f32<HH><<<waves / 8, 256, 0, stream>>>(doc, Wd, bd, pre, BB * SS, HH);
    }
    // GI = X @ W_ih^T + b_ih        (M=1280, N=768, K=256) — hoisted out of the scan
    {
        int waves = (TT * BB / 16) * (3 * HH / 64);    // 960 waves
        gemm_wmma_f32<HH><<<waves / 8, 256, 0, stream>>>(X, W_ih, b_ih, GI, TT * BB, 3 * HH);
    }

    const float* hprev = h0;
    float* hbufs[2] = { hA, hB };
    for (int t = 0; t < TT; ++t) {
        // gh = hprev @ W_hh^T + b_hh  (M=128, N=768, K=256)
        {
            int waves = (BB / 16) * (3 * HH / 64);     // 96 waves
            gemm_wmma_f32<HH><<<waves / 8, 256, 0, stream>>>(hprev, W_hh, b_hh, gh, BB, 3 * HH);
        }
        float* hnew = hbufs[t & 1];
        gru_update<<<(BB * HH + 255) / 256, 256, 0, stream>>>(GI + (size_t)t * BB * 3 * HH, gh, hprev, hnew);

        // target = hnew @ Wq^T + bq   (M=128, N=256, K=256)
        {
            int waves = (BB / 16) * (HH / 64);         // 32 waves
            gemm_wmma_f32<HH><<<waves / 8, 256, 0, stream>>>(hnew, Wq, bq, tgv, BB, HH);
        }
        // energy for step t -> out[t*B*S ..]
        energy_kernel<<<(BB * SS * 32 + 255) / 256, 256, 0, stream>>>(
            pre, tgv, Ws, bs, slen, out + (size_t)t * BB * SS);

        hprev = hnew;
    }
}